// RelGraphConv_82291573391467
// MI455X (gfx1250) — compile-verified
//
#include <hip/hip_runtime.h>

// MI455X / gfx1250, wave32. fp32 WMMA (V_WMMA_F32_16X16X4_F32) keeps the
// reference's fp32 numerics while putting the 13 GFLOP edge GEMM on the
// matrix pipe. Memory floor: ~410MB feat gather + ~410MB atomic scatter,
// both resident in the 192MB L2 (feat=25.6MB, out=25.6MB).
//
// d_ws layout:
//   [0, 128KB)      : W_rel, 8 relations, pre-swizzled WMMA-B fragment layout
//   [128KB, 144KB)  : loop_weight, same packed layout
//
// Packed B layout: element (s, h, col) = float2{ W[s*4+h*2][col],
// W[s*4+h*2+1][col] } so a lane's B fragment for k-step s is ONE b64 load:
//   bIdx = (s*2 + (lane>>4))*64 + col      (lanes 0-15 / 16-31 each 128B runs)

typedef float v2f __attribute__((ext_vector_type(2)));
typedef float v8f __attribute__((ext_vector_type(8)));

#define N_NODES 100000
#define N_EDGES 1600000
#define FEAT    64
#define NRELS   8
#define NBASES  4

#define WREL_PK_ELEMS   2048                  // float2 per 64x64 matrix
#define LWP_OFF_ELEMS   (NRELS * WREL_PK_ELEMS)

// ---------------------------------------------------------------------------
// K1a: W_rel[r] = sum_b w_comp[r,b] * weight[b], written pre-swizzled.
// 8 * 2048 float2 elements = 16384 threads.
// ---------------------------------------------------------------------------
__global__ void build_wrel_packed_kernel(const float* __restrict__ weight,  // [4,64,64]
                                         const float* __restrict__ w_comp,  // [8,4]
                                         v2f* __restrict__ wrel_pk) {
    const int idx = blockIdx.x * blockDim.x + threadIdx.x;   // 0..16383
    const int r   = idx >> 11;
    const int rem = idx & 2047;
    const int s   = rem >> 7;          // k-step 0..15
    const int h   = (rem >> 6) & 1;    // lane-half 0..1
    const int col = rem & 63;
    const int k0  = s * 4 + h * 2;

    float vx = 0.f, vy = 0.f;
#pragma unroll
    for (int b = 0; b < NBASES; ++b) {
        const float c = w_comp[r * NBASES + b];
        vx += c * weight[b * 4096 + k0 * FEAT + col];
        vy += c * weight[b * 4096 + (k0 + 1) * FEAT + col];
    }
    v2f v; v.x = vx; v.y = vy;
    wrel_pk[idx] = v;
}

// ---------------------------------------------------------------------------
// K1b: pack loop_weight into the same fragment layout (2048 float2).
// ---------------------------------------------------------------------------
__global__ void pack_loopw_kernel(const float* __restrict__ lw,   // [64,64]
                                  v2f* __restrict__ lw_pk) {
    const int idx = blockIdx.x * blockDim.x + threadIdx.x;   // 0..2047
    const int s   = idx >> 7;
    const int h   = (idx >> 6) & 1;
    const int col = idx & 63;
    const int k0  = s * 4 + h * 2;
    v2f v; v.x = lw[k0 * FEAT + col]; v.y = lw[(k0 + 1) * FEAT + col];
    lw_pk[idx] = v;
}

// ---------------------------------------------------------------------------
// K2: out = feat @ loop_weight + h_bias   (dense 100000x64x64 fp32 WMMA GEMM)
// One wave per 16x16 output tile; 16 k-steps of 16x16x4 f32 WMMA.
// Fully (re)initializes d_out before the edge scatter-adds.
// ---------------------------------------------------------------------------
__global__ void selfloop_gemm_kernel(const float* __restrict__ feat,   // [N,64]
                                     const v2f*   __restrict__ lw_pk,  // packed
                                     const float* __restrict__ bias,   // [64]
                                     float* __restrict__ out) {        // [N,64]
    const int lane  = threadIdx.x & 31;
    const int wave  = threadIdx.x >> 5;
    const int tile  = blockIdx.x * 8 + wave;   // 25000 tiles total
    const int rowT  = tile >> 2;               // 16-row tile index
    const int ntile = tile & 3;                // 16-col tile index
    const int m     = lane & 15;
    const int hh    = lane >> 4;               // 0 or 1
    const int khi   = hh * 2;
    const int ncol  = ntile * 16 + m;

    const float* arow = feat + (size_t)(rowT * 16 + m) * FEAT;

    v8f acc = {};
#pragma unroll
    for (int s = 0; s < 16; ++s) {
        const v2f a = *(const v2f*)(arow + s * 4 + khi);          // b64
        const v2f b = lw_pk[(s * 2 + hh) * FEAT + ncol];          // b64
        acc = __builtin_amdgcn_wmma_f32_16x16x4_f32(false, a, false, b,
                                                    (short)0, acc, false, false);
    }

    const float hb = bias[ncol];
#pragma unroll
    for (int j = 0; j < 8; ++j) {
        const int row = rowT * 16 + j + hh * 8;    // C/D VGPR layout
        out[(size_t)row * FEAT + ncol] = acc[j] + hb;
    }
}

// ---------------------------------------------------------------------------
// K3: edge messages + scatter.
// Each block owns 256 edges: LDS-bucket by relation, then the 8 waves consume
// 16-edge single-relation chunks. Per chunk: one k-loop keeps 4 independent
// WMMA accumulator chains (one per 16-col tile) live for matrix-pipe ILP.
// ---------------------------------------------------------------------------
__global__ void edge_msg_kernel(const float* __restrict__ feat,
                                const int*   __restrict__ src,
                                const int*   __restrict__ dst,
                                const int*   __restrict__ ety,
                                const float* __restrict__ norm,
                                const v2f*   __restrict__ wrel_pk,  // packed
                                float* __restrict__ out) {
    __shared__ int   s_src[256];
    __shared__ int   s_dst[256];
    __shared__ float s_norm[256];
    __shared__ int   s_cnt[NRELS];
    __shared__ int   s_base[NRELS];
    __shared__ int   s_chunk_rel[32];
    __shared__ int   s_chunk_start[32];
    __shared__ int   s_chunk_rows[32];
    __shared__ int   s_nchunks;

    const int tid = threadIdx.x;
    if (tid < NRELS) s_cnt[tid] = 0;
    __syncthreads();

    // ---- phase 1: load my edge, histogram by relation (LDS atomics) ----
    const int   e  = blockIdx.x * 256 + tid;        // 6250*256 == N_EDGES exact
    const int   r  = ety[e] & (NRELS - 1);
    const int   sv = src[e];
    const int   dv = dst[e];
    const float nv = norm[e];
    const int   pos = atomicAdd(&s_cnt[r], 1);
    __syncthreads();

    // ---- exclusive scan + chunk descriptors (tiny, one thread) ----
    if (tid == 0) {
        int base = 0, nch = 0;
        for (int rr = 0; rr < NRELS; ++rr) {
            s_base[rr] = base;
            const int c = s_cnt[rr];
            for (int st = 0; st < c; st += 16) {
                s_chunk_rel[nch]   = rr;
                s_chunk_start[nch] = base + st;
                s_chunk_rows[nch]  = (c - st < 16) ? (c - st) : 16;
                ++nch;
            }
            base += c;
        }
        s_nchunks = nch;
    }
    __syncthreads();

    // ---- scatter edge records into relation-sorted LDS slots ----
    const int slot = s_base[r] + pos;
    s_src[slot]  = sv;
    s_dst[slot]  = dv;
    s_norm[slot] = nv;
    __syncthreads();

    // ---- phase 2: each wave grabs chunks round-robin ----
    const int lane = tid & 31;
    const int wave = tid >> 5;
    const int m    = lane & 15;
    const int hh   = lane >> 4;
    const int khi  = hh * 2;
    const int nch  = s_nchunks;

    for (int c = wave; c < nch; c += 8) {          // wave-uniform loop
        const int rel    = s_chunk_rel[c];
        const int cstart = s_chunk_start[c];
        const int crows  = s_chunk_rows[c];
        const v2f* wr    = wrel_pk + (size_t)rel * WREL_PK_ELEMS;

        // A fragments: 16 gathered feat rows (invalid rows -> row 0, masked
        // at store time; D rows are independent so this is safe).
        const int arow_idx = (m < crows) ? s_src[cstart + m] : 0;
        const float* arow  = feat + (size_t)arow_idx * FEAT;

        v2f aF[16];
#pragma unroll
        for (int s = 0; s < 16; ++s)
            aF[s] = *(const v2f*)(arow + s * 4 + khi);            // b64

        // Per-output-row dst offset + norm, hoisted out of the tile loops.
        int   offj[8];
        float nrmj[8];
#pragma unroll
        for (int j = 0; j < 8; ++j) {
            const int rj = j + hh * 8;
            const bool ok = rj < crows;
            offj[j] = ok ? s_dst[cstart + rj] * FEAT : -1;
            nrmj[j] = ok ? s_norm[cstart + rj] : 0.f;
        }

        // 4 independent accumulator chains through a single k-loop.
        v8f acc0 = {}, acc1 = {}, acc2 = {}, acc3 = {};
#pragma unroll
        for (int s = 0; s < 16; ++s) {
            const int bBase = (s * 2 + hh) * FEAT + m;
            const v2f b0 = wr[bBase];
            const v2f b1 = wr[bBase + 16];
            const v2f b2 = wr[bBase + 32];
            const v2f b3 = wr[bBase + 48];
            acc0 = __builtin_amdgcn_wmma_f32_16x16x4_f32(false, aF[s], false, b0,
                                                         (short)0, acc0, false, false);
            acc1 = __builtin_amdgcn_wmma_f32_16x16x4_f32(false, aF[s], false, b1,
                                                         (short)0, acc1, false, false);
            acc2 = __builtin_amdgcn_wmma_f32_16x16x4_f32(false, aF[s], false, b2,
                                                         (short)0, acc2, false, false);
            acc3 = __builtin_amdgcn_wmma_f32_16x16x4_f32(false, aF[s], false, b3,
                                                         (short)0, acc3, false, false);
        }

        // Scale by norm and atomic-scatter into out[dst]. (L2-resident.)
#pragma unroll
        for (int j = 0; j < 8; ++j) {
            if (offj[j] >= 0) {
                atomicAdd(&out[offj[j] + m],      acc0[j] * nrmj[j]);
                atomicAdd(&out[offj[j] + m + 16], acc1[j] * nrmj[j]);
                atomicAdd(&out[offj[j] + m + 32], acc2[j] * nrmj[j]);
                atomicAdd(&out[offj[j] + m + 48], acc3[j] * nrmj[j]);
            }
        }
    }
}

// ---------------------------------------------------------------------------
extern "C" void kernel_launch(void* const* d_in, const int* in_sizes, int n_in,
                              void* d_out, int out_size, void* d_ws, size_t ws_size,
                              hipStream_t stream) {
    const float* feat    = (const float*)d_in[0];   // [100000,64]
    const int*   src     = (const int*)  d_in[1];   // [1600000]
    const int*   dst     = (const int*)  d_in[2];   // [1600000]
    const int*   ety     = (const int*)  d_in[3];   // [1600000]
    const float* norm    = (const float*)d_in[4];   // [1600000,1]
    const float* weight  = (const float*)d_in[5];   // [4,64,64]
    const float* w_comp  = (const float*)d_in[6];   // [8,4]
    const float* h_bias  = (const float*)d_in[7];   // [64]
    const float* loop_w  = (const float*)d_in[8];   // [64,64]
    float*       out     = (float*)d_out;           // [100000,64]

    v2f* wrel_pk = (v2f*)d_ws;                      // 128KB
    v2f* lw_pk   = (v2f*)d_ws + LWP_OFF_ELEMS;      // +16KB

    (void)in_sizes; (void)n_in; (void)out_size; (void)ws_size;

    // K1: combine bases into per-relation weights, pre-swizzled for WMMA-B.
    build_wrel_packed_kernel<<<(NRELS * WREL_PK_ELEMS) / 256, 256, 0, stream>>>(
        weight, w_comp, wrel_pk);
    pack_loopw_kernel<<<WREL_PK_ELEMS / 256, 256, 0, stream>>>(loop_w, lw_pk);

    // K2: self-loop GEMM + bias; fully re-initializes d_out every call.
    selfloop_gemm_kernel<<<3125, 256, 0, stream>>>(feat, lw_pk, h_bias, out);

    // K3: per-edge messages via LDS relation-bucketing + WMMA, atomic scatter.
    edge_msg_kernel<<<N_EDGES / 256, 256, 0, stream>>>(
        feat, src, dst, ety, norm, wrel_pk, out);
}